// TextualAttention_56238301774063
// MI455X (gfx1250) — compile-verified
//
#include <hip/hip_runtime.h>
#include <math.h>

// Problem constants (fixed by the reference)
#define BB 32
#define TT 2048
#define HH 1024
#define QQ 2048
#define NCHUNK 16            // t-chunks per batch for the fused pass
#define CT (TT / NCHUNK)     // 128 rows per chunk

typedef __attribute__((ext_vector_type(2))) float v2f;
typedef __attribute__((ext_vector_type(4))) float f4;
typedef __attribute__((ext_vector_type(8))) float v8f;

// ---------------------------------------------------------------------------
// WMMA GEMM:  C[M,N] = A[M,K] (row-major) x Bt[N,K]^T (Bt row-major over K)
// epilogue: + bias[n], optional relu, optional per-n scale (W1_w fold-in).
// One wave per 16x16 tile, K marched 4-wide with v_wmma_f32_16x16x4_f32.
// ---------------------------------------------------------------------------
__global__ __launch_bounds__(32)
void wmma_gemm_f32(const float* __restrict__ A, const float* __restrict__ Bt,
                   const float* __restrict__ bias, const float* __restrict__ nscale,
                   float* __restrict__ C, int M, int N, int K, int apply_relu) {
  const int ntiles = N >> 4;
  const int mt = blockIdx.x / ntiles;
  const int nt = blockIdx.x % ntiles;
  const int lane = threadIdx.x;        // EXEC all ones (WMMA requirement)
  const int l16  = lane & 15;
  const int koff = (lane >> 4) << 1;   // 0 or 2

  const float* arow = A  + (size_t)(mt * 16 + l16) * K + koff;
  const float* brow = Bt + (size_t)(nt * 16 + l16) * K + koff;

  v8f acc = {};
  for (int k = 0; k < K; k += 4) {
    v2f a = *(const v2f*)(arow + k);
    v2f b = *(const v2f*)(brow + k);
    acc = __builtin_amdgcn_wmma_f32_16x16x4_f32(false, a, false, b,
                                                (short)0, acc, false, false);
  }

  const int mbase = mt * 16 + ((lane >> 4) << 3);
  const int n = nt * 16 + l16;
  const float bval = bias ? bias[n] : 0.0f;
  const float sval = nscale ? nscale[n] : 1.0f;
#pragma unroll
  for (int r = 0; r < 8; ++r) {
    float v = acc[r] + bval;
    if (apply_relu) v = fmaxf(v, 0.0f);
    C[(size_t)(mbase + r) * N + n] = v * sval;
  }
}

// ---------------------------------------------------------------------------
// Fused single-pass online-softmax attention over one (batch, t-chunk).
// Each wave owns interleaved rows; a row is loaded once (non-temporal),
// scored against v[b], and folded into a running (m, s, acc) while still in
// registers. Waves combine via LDS; per-chunk (M, S, partial[H]) to global.
// ---------------------------------------------------------------------------
__global__ __launch_bounds__(256)
void fused_attention(const float* __restrict__ hidden, const float* __restrict__ v,
                     const int* __restrict__ qlen, const float* __restrict__ W1_b,
                     float* __restrict__ partial, float* __restrict__ mout,
                     float* __restrict__ sout) {
  const int b = blockIdx.x, c = blockIdx.y;
  const int tid = threadIdx.x;
  const int w = tid >> 5, lane = tid & 31;
  const int L = qlen[b];
  const int t0 = c * CT;

  __shared__ float lm[8], ls[8];
  __shared__ f4 lacc[8 * (HH / 4)];   // 32 KB

  // preload v[b] fragment: f4 index i*32+lane covers h = (i*32+lane)*4 ..+3
  const f4* v4 = (const f4*)(v + (size_t)b * HH);
  f4 vf[8];
#pragma unroll
  for (int i = 0; i < 8; ++i) vf[i] = v4[i * 32 + lane];
  const float b1 = W1_b[0];

  float m = -INFINITY, s = 0.0f;
  f4 acc[8];
#pragma unroll
  for (int i = 0; i < 8; ++i) acc[i] = (f4){0.f, 0.f, 0.f, 0.f};

  const int tend = (t0 + CT < L) ? (t0 + CT) : L;
  for (int t = t0 + w; t < tend; t += 8) {
    const f4* h4 = (const f4*)(hidden + ((size_t)b * TT + t) * HH);
    f4 row[8];
    float d = 0.0f;
#pragma unroll
    for (int i = 0; i < 8; ++i) {
      row[i] = __builtin_nontemporal_load(&h4[i * 32 + lane]);  // TH=NT stream
      d += row[i][0] * vf[i][0] + row[i][1] * vf[i][1] +
           row[i][2] * vf[i][2] + row[i][3] * vf[i][3];
    }
    for (int off = 16; off > 0; off >>= 1) d += __shfl_xor(d, off, 32);
    d += b1;                            // uniform across wave after butterfly

    float e;
    if (d > m) {                        // new max: rescale history
      const float f = (m == -INFINITY) ? 0.0f : expf(m - d);
      s *= f;
#pragma unroll
      for (int i = 0; i < 8; ++i) acc[i] *= f;
      m = d;
      e = 1.0f;                         // expf(d - m) == 1
    } else {
      e = expf(d - m);
    }
    s += e;
#pragma unroll
    for (int i = 0; i < 8; ++i) acc[i] += e * row[i];
  }

  // -------- combine the 8 waves of this block --------
  if (lane == 0) { lm[w] = m; ls[w] = s; }
  __syncthreads();
  float M = -INFINITY;
#pragma unroll
  for (int i = 0; i < 8; ++i) M = fmaxf(M, lm[i]);
  const float fw = (m == -INFINITY) ? 0.0f : expf(m - M);
#pragma unroll
  for (int i = 0; i < 8; ++i) lacc[w * (HH / 4) + i * 32 + lane] = acc[i] * fw;
  __syncthreads();

  f4 sum4 = (f4){0.f, 0.f, 0.f, 0.f};
#pragma unroll
  for (int ww = 0; ww < 8; ++ww) sum4 += lacc[ww * (HH / 4) + tid];
  ((f4*)(partial + ((size_t)b * NCHUNK + c) * HH))[tid] = sum4;

  if (tid == 0) {
    float Sb = 0.0f;
#pragma unroll
    for (int i = 0; i < 8; ++i)
      if (lm[i] != -INFINITY) Sb += ls[i] * expf(lm[i] - M);
    mout[b * NCHUNK + c] = M;
    sout[b * NCHUNK + c] = Sb;
  }
}

// ---------------------------------------------------------------------------
// Merge the NCHUNK per-chunk partials per batch (log-sum-exp combine).
// ---------------------------------------------------------------------------
__global__ __launch_bounds__(256)
void finalize_kernel(const float* __restrict__ partial, const float* __restrict__ mchunk,
                     const float* __restrict__ schunk, float* __restrict__ out) {
  const int b = blockIdx.x, tid = threadIdx.x;     // 256 threads x f4 = 1024 h
  float M = -INFINITY;
#pragma unroll
  for (int c = 0; c < NCHUNK; ++c) M = fmaxf(M, mchunk[b * NCHUNK + c]);
  float denom = 0.0f;
#pragma unroll
  for (int c = 0; c < NCHUNK; ++c) {
    const float mc = mchunk[b * NCHUNK + c];
    if (mc != -INFINITY) denom += schunk[b * NCHUNK + c] * expf(mc - M);
  }
  const float inv = (denom > 0.0f) ? 1.0f / denom : 0.0f;  // all-masked -> 0

  f4 acc = (f4){0.f, 0.f, 0.f, 0.f};
#pragma unroll
  for (int c = 0; c < NCHUNK; ++c) {
    const float mc = mchunk[b * NCHUNK + c];
    if (mc == -INFINITY) continue;
    const float fc = expf(mc - M) * inv;
    acc += fc * ((const f4*)(partial + ((size_t)b * NCHUNK + c) * HH))[tid];
  }
  ((f4*)(out + (size_t)b * HH))[tid] = acc;
}

// ---------------------------------------------------------------------------
extern "C" void kernel_launch(void* const* d_in, const int* in_sizes, int n_in,
                              void* d_out, int out_size, void* d_ws, size_t ws_size,
                              hipStream_t stream) {
  const float* hidden  = (const float*)d_in[0];  // (B,T,H)
  const float* summary = (const float*)d_in[1];  // (B,Q)
  const int*   qlen    = (const int*)d_in[2];    // (B,)
  const float* W1_w    = (const float*)d_in[3];  // (1,H)
  const float* W1_b    = (const float*)d_in[4];  // (1,)
  const float* W2_w    = (const float*)d_in[5];  // (H,Q)
  const float* W2_b    = (const float*)d_in[6];  // (H,)
  const float* W3_w    = (const float*)d_in[7];  // (Q,Q)
  const float* W3_b    = (const float*)d_in[8];  // (Q,)
  float* out = (float*)d_out;                    // (B,H)

  // workspace carve-up (floats; all partitions 16B-aligned)
  float* ws      = (float*)d_ws;
  float* s3      = ws;                              // B*Q        = 65536
  float* v       = s3 + (size_t)BB * QQ;            // B*H        = 32768
  float* partial = v + (size_t)BB * HH;             // B*NCHUNK*H = 524288 (2 MB)
  float* mchunk  = partial + (size_t)BB * NCHUNK * HH;  // B*NCHUNK = 512
  float* schunk  = mchunk + (size_t)BB * NCHUNK;        // B*NCHUNK = 512

  // 1) s3 = relu(summary @ W3^T + W3_b)             [WMMA f32 16x16x4]
  wmma_gemm_f32<<<dim3((BB / 16) * (QQ / 16)), dim3(32), 0, stream>>>(
      summary, W3_w, W3_b, nullptr, s3, BB, QQ, QQ, /*relu=*/1);

  // 2) v = (s3 @ W2^T + W2_b) * W1_w                [WMMA f32 16x16x4]
  wmma_gemm_f32<<<dim3((BB / 16) * (HH / 16)), dim3(32), 0, stream>>>(
      s3, W2_w, W2_b, W1_w, v, BB, HH, QQ, /*relu=*/0);

  // 3) single-pass fused score+softmax+weighted-sum  [one HBM pass over hidden]
  fused_attention<<<dim3(BB, NCHUNK), dim3(256), 0, stream>>>(
      hidden, v, qlen, W1_b, partial, mchunk, schunk);

  // 4) merge chunks per batch
  finalize_kernel<<<dim3(BB), dim3(256), 0, stream>>>(partial, mchunk, schunk, out);

  (void)in_sizes; (void)n_in; (void)out_size; (void)ws_size;
}